// ClassicalDDPMScheduler_63745904607917
// MI455X (gfx1250) — compile-verified
//
#include <hip/hip_runtime.h>
#include <hip/hip_bf16.h>
#include <stdint.h>

#define DDPM_T 1000

typedef float f32x4 __attribute__((ext_vector_type(4)));

// ---------------------------------------------------------------------------
// Kernel 1: build table[i] = exp( sum_{s=1..i} log1p(-beta_s) ), i = 0..1000.
// One 1024-thread block; Hillis-Steele inclusive scan in LDS.
// ---------------------------------------------------------------------------
__global__ __launch_bounds__(1024) void ddpm_build_table(float* __restrict__ lut_g) {
    __shared__ float s[1024];
    const int i = threadIdx.x;

    float x = 0.0f;
    if (i >= 1 && i <= DDPM_T) {
        // frac = (s-1)/(T-1), beta = 1e-4*(1-frac) + (20/1000)*frac
        float frac = (float)(i - 1) * (1.0f / 999.0f);
        float beta = 1e-4f * (1.0f - frac) + 0.02f * frac;
        x = log1pf(-beta);
    }
    s[i] = x;
    __syncthreads();

    // inclusive scan over 1024 entries (entry 0 stays 0 -> empty product)
    #pragma unroll
    for (int off = 1; off < 1024; off <<= 1) {
        float v = (i >= off) ? s[i - off] : 0.0f;
        __syncthreads();
        s[i] += v;
        __syncthreads();
    }

    if (i <= DDPM_T) lut_g[i] = expf(s[i]);
}

// ---------------------------------------------------------------------------
// Kernel 2: gather. Table (4004 B) is DMA'd into LDS by the Tensor Data Mover
// (one tensor_load_to_lds per block, issued by wave 0, completed with
// s_wait_tensorcnt 0, published by the workgroup barrier). Hot loop: 8
// independent non-temporal b128 loads in flight -> v_rndne -> clamp ->
// ds gather -> non-temporal b128 stores.
// ---------------------------------------------------------------------------
#if defined(__AMDGCN__) && defined(__has_builtin)
#if __has_builtin(__builtin_amdgcn_tensor_load_to_lds) && __has_builtin(__builtin_amdgcn_s_wait_tensorcnt)
#define DDPM_USE_TDM 1
#endif
#endif

#ifdef DDPM_USE_TDM
typedef __attribute__((ext_vector_type(4))) unsigned int u32x4;
typedef __attribute__((ext_vector_type(4))) int          i32x4;
typedef __attribute__((ext_vector_type(8))) int          i32x8;
#endif

#define DDPM_VPT 8   // float4 elements per thread

__global__ __launch_bounds__(256) void ddpm_gather(const f32x4* __restrict__ t4,
                                                   f32x4* __restrict__ o4,
                                                   const float* __restrict__ lut_g,
                                                   int n4, int n_rem, const float* __restrict__ t_tail,
                                                   float* __restrict__ o_tail) {
    __shared__ float lut[1024];

#ifdef DDPM_USE_TDM
    // Wave 0 issues one TDM load of the whole table: global lut_g -> LDS lut.
    if ((threadIdx.x >> 5) == 0) {
        const unsigned long long ga  = (unsigned long long)(uintptr_t)lut_g;
        const unsigned int lds_off   = (unsigned int)(uintptr_t)&lut[0]; // LDS aperture: offset in addr[31:0]

        // D# group 0 (ISA 08 §8.3): count=1 | lds_addr | global_addr[56:0] | type=2
        u32x4 g0;
        g0[0] = 1u;                                           // count=1 (valid), no gather, not restore
        g0[1] = lds_off;                                      // LDS byte address
        g0[2] = (unsigned int)(ga & 0xFFFFFFFFull);           // global_addr[31:0]
        g0[3] = (unsigned int)((ga >> 32) & 0x01FFFFFFull)    // global_addr[56:32]
              | 0x80000000u;                                  // type=2 ("image") in bits[127:126]

        // D# group 1 (§8.4): 2-D tensor, data_size=4B, dim0=tile0=1001, dim1=tile1=1
        i32x8 g1;
        g1[0] = (2 << 16);          // workgroup_mask=0, data_size=2 (4 bytes), no pad/iterate
        g1[1] = (DDPM_T + 1) << 16; // atomic_barrier_addr=0 | tensor_dim0[15:0]=1001
        g1[2] = (1 << 16);          // tensor_dim0[31:16]=0 | tensor_dim1[15:0]=1
        g1[3] = (DDPM_T + 1) << 16; // tensor_dim1[31:16]=0 | tile_dim0=1001
        g1[4] = 1;                  // tile_dim1=1 | tile_dim2=0 (unused)
        g1[5] = (DDPM_T + 1);       // tensor_dim0_stride[31:0]=1001
        g1[6] = (DDPM_T + 1) << 16; // stride0[47:32]=0 | tensor_dim1_stride[15:0]=1001
        g1[7] = 0;                  // tensor_dim1_stride[47:16]=0

        i32x4 g2 = {0, 0, 0, 0};    // dims 2/3 unused
        i32x4 g3 = {0, 0, 0, 0};    // dims 3/4 unused

#if __clang_major__ >= 23
        i32x8 g4 = {0, 0, 0, 0, 0, 0, 0, 0};
        __builtin_amdgcn_tensor_load_to_lds(g0, g1, g2, g3, g4, 0);
#else
        __builtin_amdgcn_tensor_load_to_lds(g0, g1, g2, g3, 0);
#endif
        __builtin_amdgcn_s_wait_tensorcnt(0);
    }
#else
    // Fallback: cooperative copy of the table into LDS.
    for (int k = threadIdx.x; k <= DDPM_T; k += blockDim.x) lut[k] = lut_g[k];
#endif
    __syncthreads();

    const int tid = (int)(blockIdx.x * blockDim.x + threadIdx.x);
    const int gsz = (int)(gridDim.x * blockDim.x);

    // Each thread owns DDPM_VPT float4s spaced by gsz; fully unrolled so all
    // loads issue before any consumer (8 b128 loads in flight per thread).
    f32x4 v[DDPM_VPT];
    #pragma unroll
    for (int k = 0; k < DDPM_VPT; ++k) {
        int i = tid + k * gsz;
        if (i < n4) v[k] = __builtin_nontemporal_load(&t4[i]);
    }

    #pragma unroll
    for (int k = 0; k < DDPM_VPT; ++k) {
        int i = tid + k * gsz;
        if (i < n4) {
            // jnp.round == round-half-to-even == v_rndne_f32
            int jx = (int)__builtin_rintf(v[k].x);
            int jy = (int)__builtin_rintf(v[k].y);
            int jz = (int)__builtin_rintf(v[k].z);
            int jw = (int)__builtin_rintf(v[k].w);
            jx = jx < 0 ? 0 : (jx > DDPM_T ? DDPM_T : jx);
            jy = jy < 0 ? 0 : (jy > DDPM_T ? DDPM_T : jy);
            jz = jz < 0 ? 0 : (jz > DDPM_T ? DDPM_T : jz);
            jw = jw < 0 ? 0 : (jw > DDPM_T ? DDPM_T : jw);
            f32x4 r;
            r.x = lut[jx];
            r.y = lut[jy];
            r.z = lut[jz];
            r.w = lut[jw];
            __builtin_nontemporal_store(r, &o4[i]);
        }
    }

    // Scalar tail (n not divisible by 4): at most 3 elements, thread 0..2 of block 0.
    if (n_rem && blockIdx.x == 0 && (int)threadIdx.x < n_rem) {
        float tv = t_tail[threadIdx.x];
        int j = (int)__builtin_rintf(tv);
        j = j < 0 ? 0 : (j > DDPM_T ? DDPM_T : j);
        o_tail[threadIdx.x] = lut[j];
    }
}

// ---------------------------------------------------------------------------
// Launch
// ---------------------------------------------------------------------------
extern "C" void kernel_launch(void* const* d_in, const int* in_sizes, int n_in,
                              void* d_out, int out_size, void* d_ws, size_t ws_size,
                              hipStream_t stream) {
    const float* t  = (const float*)d_in[0];
    float* out      = (float*)d_out;
    float* lut_g    = (float*)d_ws;           // 1001 floats of scratch
    const int n     = in_sizes[0];            // 16,777,216
    const int n4    = n / 4;
    const int n_rem = n - n4 * 4;

    ddpm_build_table<<<1, 1024, 0, stream>>>(lut_g);

    const int threads = 256;
    long long work = (long long)n4;
    int blocks = (int)((work + (long long)threads * DDPM_VPT - 1) / ((long long)threads * DDPM_VPT));
    if (blocks < 1) blocks = 1;
    // n = 16,777,216 -> n4 = 4,194,304 -> blocks = 2048, each thread exactly 8 float4s.
    ddpm_gather<<<blocks, threads, 0, stream>>>((const f32x4*)t, (f32x4*)out, lut_g,
                                                n4, n_rem, t + (long long)n4 * 4,
                                                out + (long long)n4 * 4);
}